// SAM_8761733283935
// MI455X (gfx1250) — compile-verified
//
#include <hip/hip_runtime.h>
#include <hip/hip_bf16.h>

typedef float v2f __attribute__((ext_vector_type(2)));
typedef float v8f __attribute__((ext_vector_type(8)));

namespace {
constexpr int Bn   = 4;
constexpr int Cin  = 256;
constexpr int Hd   = 128;
constexpr int Wd   = 128;
constexpr int HW   = Hd * Wd;        // 16384
constexpr int REL  = 32;
constexpr int OUTC = 256;
constexpr int Gn   = 32;             // OUTC / S
constexpr int MID  = 320;            // REL * 10
constexpr float EPS = 1e-5f;
}

// ---------------------------------------------------------------------------
// Generic 1x1-conv GEMM:  Y[b][o][n] = sum_c Wt[o][c] * X[b][c][n] + bias[o]
// One wave computes a 16(M) x 64(N) strip using V_WMMA_F32_16X16X4_F32.
// A frag: M = lane&15 across lanes, K = vgpr + 2*(lane>>4)
// B frag: N = lane&15 across lanes, K = vgpr + 2*(lane>>4)
// C/D:    N = lane&15,              M = vgpr + 8*(lane>>4)
// ---------------------------------------------------------------------------
__global__ __launch_bounds__(32)
void sam_gemm16(const float* __restrict__ X, const float* __restrict__ Wt,
                const float* __restrict__ bias, float* __restrict__ Y,
                int C, int O) {
  const int lane = threadIdx.x & 31;
  const int half = lane >> 4;
  const int lo   = lane & 15;
  const int n0   = blockIdx.x * 64;
  const int o0   = blockIdx.y * 16;
  const int b    = blockIdx.z;

  const float* __restrict__ Xb = X + (size_t)b * C * HW;
  float* __restrict__       Yb = Y + (size_t)b * O * HW;

  v8f acc0 = {}, acc1 = {}, acc2 = {}, acc3 = {};
  const float* __restrict__ Arow = Wt + (size_t)(o0 + lo) * C;

  for (int c = 0; c < C; c += 4) {
    v2f a = *(const v2f*)(Arow + c + 2 * half);
    const float* xr = Xb + (size_t)(c + 2 * half) * HW + n0 + lo;
    if (c + 4 < C) __builtin_prefetch(xr + (size_t)4 * HW, 0, 0);
    v2f b0, b1, b2, b3;
    b0.x = xr[0];       b0.y = xr[HW];
    b1.x = xr[16];      b1.y = xr[HW + 16];
    b2.x = xr[32];      b2.y = xr[HW + 32];
    b3.x = xr[48];      b3.y = xr[HW + 48];
    acc0 = __builtin_amdgcn_wmma_f32_16x16x4_f32(false, a, false, b0, (short)0, acc0, false, false);
    acc1 = __builtin_amdgcn_wmma_f32_16x16x4_f32(false, a, false, b1, (short)0, acc1, false, false);
    acc2 = __builtin_amdgcn_wmma_f32_16x16x4_f32(false, a, false, b2, (short)0, acc2, false, false);
    acc3 = __builtin_amdgcn_wmma_f32_16x16x4_f32(false, a, false, b3, (short)0, acc3, false, false);
  }

#pragma unroll
  for (int v = 0; v < 8; ++v) {
    const int och = o0 + v + 8 * half;
    const float bb = bias[och];
    float* yr = Yb + (size_t)och * HW + n0 + lo;
    yr[0]  = acc0[v] + bb;
    yr[16] = acc1[v] + bb;
    yr[32] = acc2[v] + bb;
    yr[48] = acc3[v] + bb;
  }
}

// ---------------------------------------------------------------------------
// Fused weight-generator + aggregation.
// One 256-thread block (8 wave32) handles 16 consecutive pixels (one tile
// stays within a single image row since W=128 and tiles are 16-aligned).
//   h[320x16]  = BN1+ReLU of concat(y1, reflect-patches of y2)   (LDS)
//   g[32x16]   = BN2+ReLU( cw1[32x320] x h )                     (WMMA, 2 waves)
//   watt[288x16] = cw2[288x32] x g + cb2                          (WMMA, 8 waves)
//   out[ch][n] = sum_k y3[ch][nbr_k(n)] * watt[(ch/8)*9 + k][n]
// ---------------------------------------------------------------------------
__global__ __launch_bounds__(256)
void sam_gen_agg(const float* __restrict__ y1, const float* __restrict__ y2,
                 const float* __restrict__ y3,
                 const float* __restrict__ bn1g, const float* __restrict__ bn1b,
                 const float* __restrict__ bn1m, const float* __restrict__ bn1v,
                 const float* __restrict__ cw1,
                 const float* __restrict__ bn2g, const float* __restrict__ bn2b,
                 const float* __restrict__ bn2m, const float* __restrict__ bn2v,
                 const float* __restrict__ cw2, const float* __restrict__ cb2,
                 float* __restrict__ out) {
  constexpr int HS = 17;                 // padded LDS row stride
  __shared__ float hL[MID * HS];         // 320 x 16 (padded)
  __shared__ float gL[32 * HS];          // 32 x 16 (padded)
  __shared__ float wL[288 * 16];         // 288 x 16

  const int tid  = threadIdx.x;
  const int wave = tid >> 5;
  const int lane = tid & 31;
  const int half = lane >> 4;
  const int lo   = lane & 15;

  const int tileIdx = blockIdx.x;            // 0 .. B*HW/16 - 1
  const int b   = tileIdx / (HW / 16);
  const int n0  = (tileIdx % (HW / 16)) * 16;
  const int hrow = n0 >> 7;                  // n0 / W
  const int w0   = n0 & (Wd - 1);

  const float* __restrict__ y1b = y1 + (size_t)b * REL * HW;
  const float* __restrict__ y2b = y2 + (size_t)b * REL * HW;
  const float* __restrict__ y3b = y3 + (size_t)b * OUTC * HW;

  // ---- build h with BN1 + ReLU ----
  for (int e = tid; e < MID * 16; e += 256) {
    const int ch = e >> 4, p = e & 15;
    float val;
    if (ch < REL) {
      val = y1b[(size_t)ch * HW + n0 + p];
    } else {
      const int cc = (ch - REL) / 9, k = (ch - REL) % 9;
      const int di = k / 3 - 1, dj = k % 3 - 1;
      int hh = hrow + di; hh = hh < 0 ? -hh : (hh >= Hd ? 2 * Hd - 2 - hh : hh);
      int ww = w0 + p + dj; ww = ww < 0 ? -ww : (ww >= Wd ? 2 * Wd - 2 - ww : ww);
      val = y2b[(size_t)cc * HW + hh * Wd + ww];
    }
    const float s = bn1g[ch] / sqrtf(bn1v[ch] + EPS);
    const float t = bn1b[ch] - bn1m[ch] * s;
    hL[ch * HS + p] = fmaxf(val * s + t, 0.0f);
  }
  __syncthreads();

  // ---- g = ReLU(BN2(cw1 x h)) : M=32 (2 tiles over waves 0,1), K=320, N=16 ----
  if (wave < 2) {
    v8f acc = {};
    const int m0 = wave * 16;
    const float* __restrict__ Arow = cw1 + (size_t)(m0 + lo) * MID;
    for (int c = 0; c < MID; c += 4) {
      v2f a = *(const v2f*)(Arow + c + 2 * half);
      v2f bf;
      bf.x = hL[(c + 2 * half) * HS + lo];
      bf.y = hL[(c + 2 * half + 1) * HS + lo];
      acc = __builtin_amdgcn_wmma_f32_16x16x4_f32(false, a, false, bf, (short)0, acc, false, false);
    }
#pragma unroll
    for (int v = 0; v < 8; ++v) {
      const int ch = m0 + v + 8 * half;
      const float s = bn2g[ch] / sqrtf(bn2v[ch] + EPS);
      const float t = bn2b[ch] - bn2m[ch] * s;
      gL[ch * HS + lo] = fmaxf(acc[v] * s + t, 0.0f);
    }
  }
  __syncthreads();

  // ---- watt = cw2 x g + cb2 : M=288 (18 tiles over 8 waves), K=32, N=16 ----
  for (int mt = wave; mt < 18; mt += 8) {
    v8f acc = {};
    const int m0 = mt * 16;
    const float* __restrict__ Arow = cw2 + (size_t)(m0 + lo) * Gn;
#pragma unroll
    for (int c = 0; c < Gn; c += 4) {
      v2f a = *(const v2f*)(Arow + c + 2 * half);
      v2f bf;
      bf.x = gL[(c + 2 * half) * HS + lo];
      bf.y = gL[(c + 2 * half + 1) * HS + lo];
      acc = __builtin_amdgcn_wmma_f32_16x16x4_f32(false, a, false, bf, (short)0, acc, false, false);
    }
#pragma unroll
    for (int v = 0; v < 8; ++v) {
      const int r = m0 + v + 8 * half;
      wL[r * 16 + lo] = acc[v] + cb2[r];
    }
  }
  __syncthreads();

  // ---- aggregation: out[ch][n0+p] = sum_k y3[ch][nbr_k] * watt[(ch>>3)*9+k][p]
  const int p  = tid & 15;
  const int c0 = tid >> 4;
  for (int i = 0; i < 16; ++i) {
    const int ch = c0 * 16 + i;              // 0..255
    const int g  = ch >> 3;
    const float* __restrict__ y3c = y3b + (size_t)ch * HW;
    float accv = 0.0f;
#pragma unroll
    for (int k = 0; k < 9; ++k) {
      const int di = k / 3 - 1, dj = k % 3 - 1;
      int hh = hrow + di; hh = hh < 0 ? -hh : (hh >= Hd ? 2 * Hd - 2 - hh : hh);
      int ww = w0 + p + dj; ww = ww < 0 ? -ww : (ww >= Wd ? 2 * Wd - 2 - ww : ww);
      accv += y3c[hh * Wd + ww] * wL[(g * 9 + k) * 16 + p];
    }
    out[((size_t)b * OUTC + ch) * HW + n0 + p] = accv;
  }
}

// ---------------------------------------------------------------------------
extern "C" void kernel_launch(void* const* d_in, const int* in_sizes, int n_in,
                              void* d_out, int out_size, void* d_ws, size_t ws_size,
                              hipStream_t stream) {
  const float* x_ref = (const float*)d_in[0];
  const float* x_cur = (const float*)d_in[1];
  const float* w1   = (const float*)d_in[2];
  const float* b1   = (const float*)d_in[3];
  const float* w2   = (const float*)d_in[4];
  const float* b2   = (const float*)d_in[5];
  const float* w3   = (const float*)d_in[6];
  const float* b3   = (const float*)d_in[7];
  const float* bn1g = (const float*)d_in[8];
  const float* bn1b = (const float*)d_in[9];
  const float* bn1m = (const float*)d_in[10];
  const float* bn1v = (const float*)d_in[11];
  const float* cw1  = (const float*)d_in[12];
  const float* bn2g = (const float*)d_in[13];
  const float* bn2b = (const float*)d_in[14];
  const float* bn2m = (const float*)d_in[15];
  const float* bn2v = (const float*)d_in[16];
  const float* cw2  = (const float*)d_in[17];
  const float* cb2  = (const float*)d_in[18];
  float* out = (float*)d_out;

  float* ws = (float*)d_ws;
  float* y1 = ws;                                     // B*REL*HW  floats (8 MB)
  float* y2 = y1 + (size_t)Bn * REL * HW;             // B*REL*HW  floats (8 MB)
  float* y3 = y2 + (size_t)Bn * REL * HW;             // B*OUTC*HW floats (64 MB)

  // 1x1 conv GEMMs (WMMA f32 16x16x4), pixels as N dimension
  sam_gemm16<<<dim3(HW / 64, REL / 16, Bn), 32, 0, stream>>>(x_cur, w1, b1, y1, Cin, REL);
  sam_gemm16<<<dim3(HW / 64, REL / 16, Bn), 32, 0, stream>>>(x_ref, w2, b2, y2, Cin, REL);
  sam_gemm16<<<dim3(HW / 64, OUTC / 16, Bn), 32, 0, stream>>>(x_cur, w3, b3, y3, Cin, OUTC);

  // fused weight-generator + aggregation
  sam_gen_agg<<<dim3(Bn * HW / 16), 256, 0, stream>>>(
      y1, y2, y3, bn1g, bn1b, bn1m, bn1v, cw1,
      bn2g, bn2b, bn2m, bn2v, cw2, cb2, out);
}